// VectorQuantizer_24996709662906
// MI455X (gfx1250) — compile-verified
//
#include <hip/hip_runtime.h>
#include <stdint.h>

#define B_ 8
#define T_ 4096
#define D_ 64
#define K_ 8192
#define N_ (B_ * T_)

typedef float v2f __attribute__((ext_vector_type(2)));
typedef float v8f __attribute__((ext_vector_type(8)));

// pointer types matching the async-LDS builtin's expected parameters
typedef int v4i_ __attribute__((vector_size(16)));
typedef __attribute__((address_space(1))) v4i_* gv4i_ptr;   // global
typedef __attribute__((address_space(3))) v4i_* lv4i_ptr;   // LDS

#if __has_builtin(__builtin_amdgcn_global_load_async_to_lds_b128)
#define VQ_ASYNC_LDS 1
#else
#define VQ_ASYNC_LDS 0
#endif

// ---------------------------------------------------------------------------
// Pass 1: csq[k] = 0.5 * ||codebook[k]||^2
// ---------------------------------------------------------------------------
__global__ __launch_bounds__(256) void vq_csq_kernel(const float* __restrict__ cb,
                                                     float* __restrict__ csq) {
  int k = blockIdx.x * 256 + threadIdx.x;
  const float4* row = reinterpret_cast<const float4*>(cb + (size_t)k * D_);
  float s = 0.f;
#pragma unroll
  for (int i = 0; i < D_ / 4; ++i) {
    float4 q = row[i];
    s += q.x * q.x + q.y * q.y + q.z * q.z + q.w * q.w;
  }
  csq[k] = 0.5f * s;
}

// ---------------------------------------------------------------------------
// Pass 2: argmin_k( 0.5||c_k||^2 - x_n . c_k ) via exact fp32 WMMA.
// Wave: one 16-row strip of (-x) held as A fragments for the whole K loop.
// Block: 8 waves (128 rows); codebook double-buffered through LDS using the
// CDNA5 async global->LDS data path (ASYNCcnt).
// ---------------------------------------------------------------------------
constexpr int CHUNK  = 64;          // codes per LDS chunk
constexpr int RP     = 68;          // padded row pitch in floats (272B, 16B aligned)
constexpr int NCHUNK = K_ / CHUNK;  // 128
constexpr int TILES  = CHUNK / 16;  // 4

__device__ __forceinline__ void vq_wait_async() {
#if VQ_ASYNC_LDS
#if __has_builtin(__builtin_amdgcn_s_wait_asynccnt)
  __builtin_amdgcn_s_wait_asynccnt(0);
#else
  asm volatile("s_wait_asynccnt 0x0" ::: "memory");
#endif
#endif
}

// Copy one 64-code chunk (16 KB) global -> LDS with row padding.
// Async path: 4 x b128 async-LDS loads per thread, zero VGPR data movement.
__device__ __forceinline__ void vq_stage_chunk(const float* __restrict__ cb,
                                               float* __restrict__ lbuf,
                                               int c, int tid) {
  const char* gbase = reinterpret_cast<const char*>(cb + (size_t)c * CHUNK * D_);
#pragma unroll
  for (int i = 0; i < 4; ++i) {
    int f    = tid + 256 * i;   // float4 index within chunk (64 codes * 16)
    int code = f >> 4;
    int d4   = f & 15;
    const char* g = gbase + (size_t)f * 16;
    float*      l = lbuf + code * RP + d4 * 4;
#if VQ_ASYNC_LDS
    __builtin_amdgcn_global_load_async_to_lds_b128(
        (gv4i_ptr)(uintptr_t)g,
        (lv4i_ptr)(uint32_t)(uintptr_t)l,
        /*offset=*/0, /*cpol=*/0);
#else
    *reinterpret_cast<float4*>(l) = *reinterpret_cast<const float4*>(g);
#endif
  }
}

__global__ __launch_bounds__(256) void vq_argmin_kernel(const float* __restrict__ x,
                                                        const float* __restrict__ cb,
                                                        const float* __restrict__ csq,
                                                        int* __restrict__ out) {
  __shared__ float lds[2][CHUNK * RP];

  const int tid  = threadIdx.x;
  const int wave = tid >> 5;
  const int lane = tid & 31;
  const int r    = lane & 15;  // A: row M; B/C: column N (code-in-tile)
  const int h    = lane >> 4;  // half select (K pair for A/B; +8 rows for C/D)

  const int rowBase = blockIdx.x * 128 + wave * 16;

  // ---- A fragments: 16x64 strip of (-x); 32-bit 16x4 A layout per j-chunk ----
  v2f a[16];
  {
    const float* xrow = x + (size_t)(rowBase + r) * D_ + 2 * h;
#pragma unroll
    for (int j = 0; j < 16; ++j) {
      v2f t = *reinterpret_cast<const v2f*>(xrow + 4 * j);
      a[j] = -t;  // D = C + (-x).c = csq - dot
    }
  }

  float bestv[8];
  int   besti[8];
#pragma unroll
  for (int v = 0; v < 8; ++v) { bestv[v] = 3.4e38f; besti[v] = 0; }

  vq_stage_chunk(cb, lds[0], 0, tid);

  for (int c = 0; c < NCHUNK; ++c) {
    vq_wait_async();   // our async copies into buf c&1 are committed to LDS
    __syncthreads();   // everyone's copies visible; prior readers of buf (c+1)&1 done
    if (c + 1 < NCHUNK) vq_stage_chunk(cb, lds[(c + 1) & 1], c + 1, tid);

    const float* buf = lds[c & 1];
    const int    kc  = c * CHUNK;
    float cs[TILES];
#pragma unroll
    for (int t = 0; t < TILES; ++t) cs[t] = csq[kc + t * 16 + r];

#pragma unroll
    for (int t = 0; t < TILES; ++t) {
      const float* brow = buf + (t * 16 + r) * RP + 2 * h;

      // preload the whole tile's B fragments first: 8 back-to-back
      // ds_load_2addr_b64, one dscnt wait, then an uninterrupted WMMA chain
      v2f b[16];
#pragma unroll
      for (int j = 0; j < 16; ++j) b[j] = *reinterpret_cast<const v2f*>(brow + 4 * j);

      v8f acc = {cs[t], cs[t], cs[t], cs[t], cs[t], cs[t], cs[t], cs[t]};
#pragma unroll
      for (int j = 0; j < 16; ++j) {
        acc = __builtin_amdgcn_wmma_f32_16x16x4_f32(false, a[j], false, b[j],
                                                    (short)0, acc, false, false);
      }

      const int code = kc + t * 16 + r;
#pragma unroll
      for (int v = 0; v < 8; ++v) {
        if (acc[v] < bestv[v]) { bestv[v] = acc[v]; besti[v] = code; }
      }
    }
  }

  // ---- cross-lane argmin over the 16 columns within each half ----
#pragma unroll
  for (int m = 8; m >= 1; m >>= 1) {
#pragma unroll
    for (int v = 0; v < 8; ++v) {
      float ov = __shfl_xor(bestv[v], m, 32);
      int   oi = __shfl_xor(besti[v], m, 32);
      if (ov < bestv[v] || (ov == bestv[v] && oi < besti[v])) {
        bestv[v] = ov;
        besti[v] = oi;
      }
    }
  }
  if (r == 0) {  // lane 0 -> rows 0..7 of strip, lane 16 -> rows 8..15
#pragma unroll
    for (int v = 0; v < 8; ++v) out[rowBase + h * 8 + v] = besti[v];
  }
}

// ---------------------------------------------------------------------------
extern "C" void kernel_launch(void* const* d_in, const int* in_sizes, int n_in,
                              void* d_out, int out_size, void* d_ws, size_t ws_size,
                              hipStream_t stream) {
  const float* x   = (const float*)d_in[0];
  const float* cb  = (const float*)d_in[1];
  float*       csq = (float*)d_ws;  // 8192 floats = 32 KB scratch
  int*         out = (int*)d_out;   // reference output dtype: int32 indices

  vq_csq_kernel<<<K_ / 256, 256, 0, stream>>>(cb, csq);
  vq_argmin_kernel<<<N_ / 128, 256, 0, stream>>>(x, cb, csq, out);
}